// GCN_45853070852446
// MI455X (gfx1250) — compile-verified
//
#include <hip/hip_runtime.h>
#include <math.h>

typedef __attribute__((ext_vector_type(2))) float v2f;
typedef __attribute__((ext_vector_type(8))) float v8f;

// ---------------------------------------------------------------------------
// degree / normalization
// ---------------------------------------------------------------------------
__global__ void zero_f32(float* __restrict__ p, int n) {
    int i = blockIdx.x * blockDim.x + threadIdx.x;
    if (i < n) p[i] = 0.0f;
}

__global__ void degree_kernel(const int* __restrict__ dst, float* __restrict__ deg, int E) {
    int e = blockIdx.x * blockDim.x + threadIdx.x;
    if (e < E) atomicAdd(&deg[dst[e]], 1.0f);
}

// dinv[i] = rsqrt(deg_edges[i] + 1)   (+1 == self loop; always > 0)
__global__ void dinv_kernel(float* __restrict__ deg, int n) {
    int i = blockIdx.x * blockDim.x + threadIdx.x;
    if (i < n) deg[i] = rsqrtf(deg[i] + 1.0f);
}

// ---------------------------------------------------------------------------
// GEMM1: h[N,16] = x[N,128] @ W1[128,16] via V_WMMA_F32_16X16X4_F32
// One wave handles one 16-row tile; 32 WMMAs (K=128 in steps of 4).
// Tail rows are CLAMPED (not predicated): garbage A rows only feed garbage
// D rows which are never stored, so all loads are unconditional b64s.
// ---------------------------------------------------------------------------
__global__ void gemm1_wmma(const float* __restrict__ x, const float* __restrict__ W1,
                           float* __restrict__ h, int nrows) {
    const int lane = threadIdx.x & 31;
    const int half = lane >> 4;       // 0: lanes 0-15, 1: lanes 16-31
    const int l    = lane & 15;
    const int wid  = blockIdx.x * (blockDim.x >> 5) + (threadIdx.x >> 5);
    const int nw   = gridDim.x * (blockDim.x >> 5);
    const int ntiles = (nrows + 15) >> 4;

    // Preload all B fragments of W1 (K=128): .x holds row 4kb+2*half,
    // .y holds row 4kb+2*half+1 (B 4x16 layout, lanes = N columns).
    v2f bfrag[32];
#pragma unroll
    for (int kb = 0; kb < 32; ++kb) {
        const float* wp = W1 + (4 * kb + 2 * half) * 16 + l;
        bfrag[kb].x = wp[0];
        bfrag[kb].y = wp[16];
    }

    for (int t = wid; t < ntiles; t += nw) {
        int rowA = 16 * t + l;                   // A-matrix row for this lane
        if (rowA >= nrows) rowA = nrows - 1;     // clamp: tail garbage is discarded
        const float* xp = x + (size_t)rowA * 128 + 2 * half;
        v8f c = {};
#pragma unroll
        for (int kb = 0; kb < 32; ++kb) {
            v2f a = *(const v2f*)(xp + 4 * kb);  // A[row][4kb+2*half .. +1], b64 load
            c = __builtin_amdgcn_wmma_f32_16x16x4_f32(
                    false, a, false, bfrag[kb], (short)0, c, false, false);
        }
#pragma unroll
        for (int r = 0; r < 8; ++r) {
            int row = 16 * t + 8 * half + r;     // D layout: c[r] = D[r+8*half][l]
            if (row < nrows) h[(size_t)row * 16 + l] = c[r];
        }
    }
}

// ---------------------------------------------------------------------------
// GEMM2: h2[N,4] = h1[N,16] @ W2[16,4]  (W2 zero-padded to 16 cols in regs)
// ---------------------------------------------------------------------------
__global__ void gemm2_wmma(const float* __restrict__ h1, const float* __restrict__ W2,
                           float* __restrict__ h2, int nrows) {
    const int lane = threadIdx.x & 31;
    const int half = lane >> 4;
    const int l    = lane & 15;
    const int wid  = blockIdx.x * (blockDim.x >> 5) + (threadIdx.x >> 5);
    const int nw   = gridDim.x * (blockDim.x >> 5);
    const int ntiles = (nrows + 15) >> 4;

    v2f bfrag[4];
#pragma unroll
    for (int kb = 0; kb < 4; ++kb) {
        int r0 = 4 * kb + 2 * half;              // K row
        bfrag[kb].x = (l < 4) ? W2[r0 * 4 + l] : 0.0f;
        bfrag[kb].y = (l < 4) ? W2[(r0 + 1) * 4 + l] : 0.0f;
    }

    for (int t = wid; t < ntiles; t += nw) {
        int rowA = 16 * t + l;
        if (rowA >= nrows) rowA = nrows - 1;     // clamp, see gemm1
        const float* hp = h1 + (size_t)rowA * 16 + 2 * half;
        v8f c = {};
#pragma unroll
        for (int kb = 0; kb < 4; ++kb) {
            v2f a = *(const v2f*)(hp + 4 * kb);
            c = __builtin_amdgcn_wmma_f32_16x16x4_f32(
                    false, a, false, bfrag[kb], (short)0, c, false, false);
        }
#pragma unroll
        for (int r = 0; r < 8; ++r) {
            int row = 16 * t + 8 * half + r;
            if (row < nrows && l < 4) h2[(size_t)row * 4 + l] = c[r];
        }
    }
}

// ---------------------------------------------------------------------------
// Scatter layer 1 (16 channels)
// ---------------------------------------------------------------------------
__global__ void init_acc16(const float* __restrict__ hin, const float* __restrict__ dinv,
                           float* __restrict__ acc, int n) {
    int i = blockIdx.x * blockDim.x + threadIdx.x;
    if (i >= n) return;
    float w = dinv[i] * dinv[i];                // self-loop norm
    const float4* hp = (const float4*)(hin + (size_t)i * 16);
    float4* ap = (float4*)(acc + (size_t)i * 16);
#pragma unroll
    for (int q = 0; q < 4; ++q) {
        float4 v = hp[q];
        ap[q] = make_float4(v.x * w, v.y * w, v.z * w, v.w * w);
    }
}

__global__ void scatter_edges16(const int* __restrict__ src, const int* __restrict__ dst,
                                const float* __restrict__ dinv, const float* __restrict__ hin,
                                float* __restrict__ acc, int E) {
    int e = blockIdx.x * blockDim.x + threadIdx.x;
    if (e >= E) return;
    int s = src[e], d = dst[e];
    float norm = dinv[s] * dinv[d];
    const float4* hp = (const float4*)(hin + (size_t)s * 16);
    float* ap = acc + (size_t)d * 16;
#pragma unroll
    for (int q = 0; q < 4; ++q) {
        float4 v = hp[q];
        atomicAdd(ap + 4 * q + 0, v.x * norm);
        atomicAdd(ap + 4 * q + 1, v.y * norm);
        atomicAdd(ap + 4 * q + 2, v.z * norm);
        atomicAdd(ap + 4 * q + 3, v.w * norm);
    }
}

__global__ void bias_relu16(float* __restrict__ acc, const float* __restrict__ b, int n) {
    int i = blockIdx.x * blockDim.x + threadIdx.x;
    if (i >= n) return;
    float* p = acc + (size_t)i * 16;
#pragma unroll
    for (int c = 0; c < 16; ++c) {
        float v = p[c] + b[c];
        p[c] = v > 0.0f ? v : 0.0f;
    }
}

// ---------------------------------------------------------------------------
// Scatter layer 2 (4 channels) -> accumulate directly into d_out
// ---------------------------------------------------------------------------
__global__ void init_out4(const float* __restrict__ h2, const float* __restrict__ dinv,
                          float* __restrict__ out, int n) {
    int i = blockIdx.x * blockDim.x + threadIdx.x;
    if (i >= n) return;
    float w = dinv[i] * dinv[i];
    float4 v = *(const float4*)(h2 + (size_t)i * 4);
    *(float4*)(out + (size_t)i * 4) = make_float4(v.x * w, v.y * w, v.z * w, v.w * w);
}

__global__ void scatter_edges4(const int* __restrict__ src, const int* __restrict__ dst,
                               const float* __restrict__ dinv, const float* __restrict__ h2,
                               float* __restrict__ out, int E) {
    int e = blockIdx.x * blockDim.x + threadIdx.x;
    if (e >= E) return;
    int s = src[e], d = dst[e];
    float norm = dinv[s] * dinv[d];
    float4 v = *(const float4*)(h2 + (size_t)s * 4);
    float* op = out + (size_t)d * 4;
    atomicAdd(op + 0, v.x * norm);
    atomicAdd(op + 1, v.y * norm);
    atomicAdd(op + 2, v.z * norm);
    atomicAdd(op + 3, v.w * norm);
}

__global__ void logsoftmax4(float* __restrict__ out, const float* __restrict__ b2, int n) {
    int i = blockIdx.x * blockDim.x + threadIdx.x;
    if (i >= n) return;
    float* p = out + (size_t)i * 4;
    float v0 = p[0] + b2[0], v1 = p[1] + b2[1], v2 = p[2] + b2[2], v3 = p[3] + b2[3];
    float m = fmaxf(fmaxf(v0, v1), fmaxf(v2, v3));
    float s = expf(v0 - m) + expf(v1 - m) + expf(v2 - m) + expf(v3 - m);
    float lse = m + logf(s);
    p[0] = v0 - lse; p[1] = v1 - lse; p[2] = v2 - lse; p[3] = v3 - lse;
}

// ---------------------------------------------------------------------------
// Orchestration
// ---------------------------------------------------------------------------
extern "C" void kernel_launch(void* const* d_in, const int* in_sizes, int n_in,
                              void* d_out, int out_size, void* d_ws, size_t ws_size,
                              hipStream_t stream) {
    const float* x   = (const float*)d_in[0];   // [N,128]
    const int*   ei  = (const int*)  d_in[1];   // [2,E] row-major
    const float* W1  = (const float*)d_in[2];   // [128,16]
    const float* b1  = (const float*)d_in[3];   // [16]
    const float* W2  = (const float*)d_in[4];   // [16,4]
    const float* b2  = (const float*)d_in[5];   // [4]
    float* out = (float*)d_out;                 // [N,4]

    const int N = in_sizes[0] / 128;
    const int E = in_sizes[1] / 2;
    const int* src = ei;
    const int* dst = ei + E;

    char* ws = (char*)d_ws;
    float* dinv = (float*)(ws);                       // N floats        (0.4 MB)
    float* htmp = (float*)(ws + (1u  << 20));         // N*16 floats     (6.4 MB)
    float* acc1 = (float*)(ws + (8u  << 20));         // N*16 floats     (6.4 MB)
    float* h2   = (float*)(ws + (15u << 20));         // N*4 floats      (1.6 MB)

    const int B = 256;
    const int gN  = (N + B - 1) / B;
    const int gE  = (E + B - 1) / B;
    const int ntiles = (N + 15) / 16;
    const int gT  = (ntiles + 7) / 8;                 // 8 waves / block

    // normalization
    zero_f32<<<gN, B, 0, stream>>>(dinv, N);
    degree_kernel<<<gE, B, 0, stream>>>(dst, dinv, E);
    dinv_kernel<<<gN, B, 0, stream>>>(dinv, N);

    // layer 1
    gemm1_wmma<<<gT, B, 0, stream>>>(x, W1, htmp, N);
    init_acc16<<<gN, B, 0, stream>>>(htmp, dinv, acc1, N);
    scatter_edges16<<<gE, B, 0, stream>>>(src, dst, dinv, htmp, acc1, E);
    bias_relu16<<<gN, B, 0, stream>>>(acc1, b1, N);

    // layer 2
    gemm2_wmma<<<gT, B, 0, stream>>>(acc1, W2, h2, N);
    init_out4<<<gN, B, 0, stream>>>(h2, dinv, out, N);
    scatter_edges4<<<gE, B, 0, stream>>>(src, dst, dinv, h2, out, E);
    logsoftmax4<<<gN, B, 0, stream>>>(out, b2, N);
}